// NequIP_FlowMatching_29953101922745
// MI455X (gfx1250) — compile-verified
//
#include <hip/hip_runtime.h>
#include <math.h>

#define N_ATOMS 4096
#define N_EDGES 49152

// ---------------------------------------------------------------------------
// Types for WMMA
// ---------------------------------------------------------------------------
typedef __attribute__((ext_vector_type(2))) float v2f;
typedef __attribute__((ext_vector_type(8))) float v8f;

// ---------------------------------------------------------------------------
// Generic strided GEMM on the f32 WMMA pipe:
//   C[m,n] (+)= act( scale * sum_k A[m,k]*B[k,n] )
// A: element (m,k) at A[m*arow + k*acol]   (CONTIG=1 -> acol==1, b64 loads)
// B: dense K x N row-major
// C: element (m,n) at C[m*crow + n*ccol]
// One wave computes one 16x16 tile with V_WMMA_F32_16X16X4_F32.
//
// NOTE: every call site uses K % 4 == 0, so there are no K-edge guards.
// M/N edges are handled by clamping load indices (out-of-range rows/cols
// only feed C entries that the store mask drops), so the inner loop is
// branch-free: 4 loads + 1 WMMA + pointer bumps.
// ---------------------------------------------------------------------------
template <int CONTIG>
__global__ __launch_bounds__(32)
void k_gemm_wmma(const float* __restrict__ A, int arow, int acol,
                 const float* __restrict__ B,
                 float* __restrict__ C, int crow, int ccol,
                 int M, int K, int N, float scale, int act, int accum)
{
    const int lane = threadIdx.x & 31;
    const int half = lane >> 4;      // 0: K pair (0,1), 1: K pair (2,3)
    const int l    = lane & 15;
    const int m0   = blockIdx.y << 4;
    const int n0   = blockIdx.x << 4;

    const int am  = m0 + l;                      // A row (ISA 16x4 layout)
    const int bn  = n0 + l;                      // B/C column
    const int amc = (am < M) ? am : (M - 1);     // clamped (fault-free)
    const int bnc = (bn < N) ? bn : (N - 1);
    const int ac  = CONTIG ? 1 : acol;

    const float* pa = A + (size_t)amc * arow + (size_t)(half << 1) * ac;
    const float* pb = B + (size_t)(half << 1) * N + bnc;
    const size_t astep = (size_t)4 * ac;
    const size_t bstep = (size_t)4 * N;

    v8f acc = {};
#pragma unroll 2
    for (int k0 = 0; k0 < K; k0 += 4) {
        v2f a, b;
        if (CONTIG) {
            a = *(const v2f*)pa;                 // aligned global_load_b64
        } else {
            a.x = pa[0];
            a.y = pa[ac];
        }
        b.x = pb[0];
        b.y = pb[N];
        acc = __builtin_amdgcn_wmma_f32_16x16x4_f32(false, a, false, b,
                                                    (short)0, acc, false, false);
        pa += astep;
        pb += bstep;
    }

    if (bn < N) {
#pragma unroll
        for (int r = 0; r < 8; ++r) {
            int cm = m0 + (half << 3) + r;       // C layout: vgpr r -> M=r / 8+r
            if (cm < M) {
                size_t off = (size_t)cm * crow + (size_t)bn * ccol;
                float v = acc[r] * scale;
                if (accum) v += C[off];
                if (act)   v = v / (1.f + expf(-v));  // SiLU
                C[off] = v;
            }
        }
    }
}

// ---------------------------------------------------------------------------
// Zero fill
// ---------------------------------------------------------------------------
__global__ void k_fill(float* __restrict__ p, long n)
{
    long i = (long)blockIdx.x * blockDim.x + threadIdx.x;
    if (i < n) p[i] = 0.f;
}

// ---------------------------------------------------------------------------
// Spherical harmonics per edge: sh[e][0..8] = [Y0 | Y1(y,z,x) | Y2(5)]
// ---------------------------------------------------------------------------
__global__ void k_sph(const float* __restrict__ ev, float* __restrict__ sh)
{
    int e = blockIdx.x * blockDim.x + threadIdx.x;
    if (e >= N_EDGES) return;
    float x0 = ev[e * 3 + 0], y0 = ev[e * 3 + 1], z0 = ev[e * 3 + 2];
    float r = sqrtf(x0 * x0 + y0 * y0 + z0 * z0);
    float inv = 1.f / (r + 1e-12f);
    float x = x0 * inv, y = y0 * inv, z = z0 * inv;
    float* s = sh + (size_t)e * 9;
    const float s3 = 1.7320508075688772f;
    const float s15 = 3.872983346207417f;
    const float s5h = 1.118033988749895f;
    const float s15h = 1.9364916731037085f;
    s[0] = 1.f;
    s[1] = s3 * y;  s[2] = s3 * z;  s[3] = s3 * x;
    s[4] = s15 * x * y;
    s[5] = s15 * y * z;
    s[6] = s5h * (3.f * z * z - 1.f);
    s[7] = s15 * x * z;
    s[8] = s15h * (x * x - y * y);
}

// ---------------------------------------------------------------------------
// Outer product P[n, u*8+v] = feat[n, base + u*fstride + m] * z[n, v]
// ---------------------------------------------------------------------------
__global__ void k_outer(const float* __restrict__ fin, int frow, int fbase,
                        int fstride, int m, const float* __restrict__ z,
                        float* __restrict__ P, int mul)
{
    int i = blockIdx.x * blockDim.x + threadIdx.x;
    int K = mul * 8;
    int tot = N_ATOMS * K;
    if (i >= tot) return;
    int n = i / K;
    int r = i - n * K;
    int u = r >> 3, v = r & 7;
    P[i] = fin[(size_t)n * frow + fbase + u * fstride + m] * z[n * 8 + v];
}

// ---------------------------------------------------------------------------
// Fused tensor-product message pass with scatter (atomic f32 adds).
// ---------------------------------------------------------------------------
struct MPath { int mul, x1b, d1, shb, d2, d3, cgoff, msgb; };
struct MTable { int npaths, nw; MPath p[15]; };

__global__ __launch_bounds__(256)
void k_msg(const int* __restrict__ ei, const float* __restrict__ x1,
           const float* __restrict__ sh, const float* __restrict__ wbuf,
           const float* __restrict__ w3j, float* __restrict__ msg, MTable T)
{
    __shared__ float scg[615];
    for (int i = threadIdx.x; i < 615; i += 256) scg[i] = w3j[i];
    __syncthreads();

    int tid = blockIdx.x * 256 + threadIdx.x;
    int e = tid / T.nw;
    if (e >= N_EDGES) return;
    int idx = tid - e * T.nw;

    int pi = 0, u = idx;
    while (u >= T.p[pi].mul) { u -= T.p[pi].mul; ++pi; }
    MPath pt = T.p[pi];

    int s = ei[e];
    int d = ei[N_EDGES + e];

    float xe[5], shv[5], outk[5];
    for (int i = 0; i < pt.d1; ++i)
        xe[i] = x1[(size_t)s * 320 + pt.x1b + u * pt.d1 + i];
    for (int j = 0; j < pt.d2; ++j)
        shv[j] = sh[(size_t)e * 9 + pt.shb + j];
    for (int k = 0; k < pt.d3; ++k) outk[k] = 0.f;

    const float* cg = scg + pt.cgoff;
    for (int i = 0; i < pt.d1; ++i)
        for (int j = 0; j < pt.d2; ++j) {
            float xs = xe[i] * shv[j];
            const float* c2 = cg + (i * pt.d2 + j) * pt.d3;
            for (int k = 0; k < pt.d3; ++k) outk[k] += c2[k] * xs;
        }

    float wv = wbuf[(size_t)e * T.nw + idx] * 0.28867513459481287f; // 1/sqrt(12)
    float* dstp = msg + (size_t)d * 1920 + pt.msgb + u * pt.d3;
    for (int k = 0; k < pt.d3; ++k) atomicAdd(dstp + k, outk[k] * wv);
}

// ---------------------------------------------------------------------------
// Gate + conditioning: feat_new(4096,384) -> feat(4096,320), += h_ty[batch]
// ---------------------------------------------------------------------------
__global__ void k_gate(const float* __restrict__ fnew, const float* __restrict__ hty,
                       const int* __restrict__ batch, float* __restrict__ feat)
{
    int i = blockIdx.x * blockDim.x + threadIdx.x;
    if (i >= N_ATOMS * 320) return;
    int n = i / 320, c = i - n * 320;
    const float* f = fnew + (size_t)n * 384;
    float v;
    if (c < 64) {
        float x = f[c];
        v = x / (1.f + expf(-x));                       // silu scalars
    } else if (c < 160) {
        int r = c - 64; int u = r / 3;
        v = f[128 + r] * (1.f / (1.f + expf(-f[64 + u])));  // gate l=1
    } else {
        int r = c - 160; int u = r / 5;
        v = f[224 + r] * (1.f / (1.f + expf(-f[96 + u])));  // gate l=2
    }
    feat[i] = v + hty[batch[n] * 320 + c];
}

// ---------------------------------------------------------------------------
// Final projection: out[n,m] = (1/16) sum_{u<32,v<8} feat1[n,u,m] z[n,v] W[u,v]
// ---------------------------------------------------------------------------
__global__ void k_final(const float* __restrict__ feat, const float* __restrict__ z,
                        const float* __restrict__ wout, float* __restrict__ out)
{
    int i = blockIdx.x * blockDim.x + threadIdx.x;
    if (i >= N_ATOMS * 3) return;
    int n = i / 3, m = i - 3 * n;
    float acc = 0.f;
    for (int u = 0; u < 32; ++u) {
        float fu = feat[(size_t)n * 320 + 64 + u * 3 + m];
        for (int v = 0; v < 8; ++v) acc += fu * z[n * 8 + v] * wout[u * 8 + v];
    }
    out[i] = acc * 0.0625f;
}

// ---------------------------------------------------------------------------
// t/y Gaussian embeddings + 2-layer MLP projections -> hty (4,320)
// ---------------------------------------------------------------------------
__global__ __launch_bounds__(320)
void k_embed(const float* __restrict__ t, const float* __restrict__ y,
             const float* tw1, const float* tb1, const float* tw2, const float* tb2,
             const float* pw1, const float* pb1, const float* pw2, const float* pb2,
             const float* yw1, const float* yb1, const float* yw2, const float* yb2,
             const float* qw1, const float* qb1, const float* qw2, const float* qb2,
             float* __restrict__ hty)
{
    __shared__ float basis[4][12];
    __shared__ float emb[4][64];
    __shared__ float g32[4][32];
    __shared__ float mid[4][320];
    int tid = threadIdx.x;

    // ---- t branch ----
    if (tid < 48) { int g = tid / 12, j = tid % 12;
        float mean = (float)j * (1.f / 11.f);
        float dv = (t[g] - mean) * 10.f;       // /0.1
        basis[g][j] = expf(-0.5f * dv * dv);
    }
    __syncthreads();
    if (tid < 256) { int g = tid >> 6, h = tid & 63;
        float a = tb1[h];
        for (int j = 0; j < 12; ++j) a += basis[g][j] * tw1[j * 64 + h];
        emb[g][h] = a / (1.f + expf(-a));
    }
    __syncthreads();
    if (tid < 128) { int g = tid >> 5, o = tid & 31;
        float a = tb2[o];
        for (int h = 0; h < 64; ++h) a += emb[g][h] * tw2[h * 32 + o];
        g32[g][o] = a;
    }
    __syncthreads();
    { int dd = tid;
        for (int g = 0; g < 4; ++g) {
            float a = pb1[dd];
            for (int o = 0; o < 32; ++o) a += g32[g][o] * pw1[o * 320 + dd];
            mid[g][dd] = a / (1.f + expf(-a));
        }
    }
    __syncthreads();
    { int dd = tid;
        for (int g = 0; g < 4; ++g) {
            float a = pb2[dd];
            for (int h = 0; h < 320; ++h) a += mid[g][h] * pw2[h * 320 + dd];
            hty[g * 320 + dd] = a;
        }
    }
    __syncthreads();

    // ---- y branch (accumulate) ----
    if (tid < 36) { int g = tid / 9, j = tid % 9;
        float mean = -1.f + 0.5f * (float)j;
        float dv = (y[g] - mean) * (1.f / 0.6f);
        basis[g][j] = expf(-0.5f * dv * dv);
    }
    __syncthreads();
    if (tid < 256) { int g = tid >> 6, h = tid & 63;
        float a = yb1[h];
        for (int j = 0; j < 9; ++j) a += basis[g][j] * yw1[j * 64 + h];
        emb[g][h] = a / (1.f + expf(-a));
    }
    __syncthreads();
    if (tid < 128) { int g = tid >> 5, o = tid & 31;
        float a = yb2[o];
        for (int h = 0; h < 64; ++h) a += emb[g][h] * yw2[h * 32 + o];
        g32[g][o] = a;
    }
    __syncthreads();
    { int dd = tid;
        for (int g = 0; g < 4; ++g) {
            float a = qb1[dd];
            for (int o = 0; o < 32; ++o) a += g32[g][o] * qw1[o * 320 + dd];
            mid[g][dd] = a / (1.f + expf(-a));
        }
    }
    __syncthreads();
    { int dd = tid;
        for (int g = 0; g < 4; ++g) {
            float a = qb2[dd];
            for (int h = 0; h < 320; ++h) a += mid[g][h] * qw2[h * 320 + dd];
            hty[g * 320 + dd] += a;
        }
    }
}

// ---------------------------------------------------------------------------
// Device-side real Wigner-3j table (translation of the reference Python).
// Single thread, double precision, runs once per launch (negligible).
// ---------------------------------------------------------------------------
struct dc { double re, im; };
__device__ inline dc cmul(dc a, dc b) { return {a.re * b.re - a.im * b.im, a.re * b.im + a.im * b.re}; }
__device__ inline dc cconj(dc a) { return {a.re, -a.im}; }

__device__ double dfact(int n) { double r = 1.0; for (int i = 2; i <= n; ++i) r *= i; return r; }

__device__ double cgc(int j1, int m1, int j2, int m2, int j3, int m3)
{
    if (m1 + m2 != m3) return 0.0;
    double pre = sqrt((double)(2 * j3 + 1) * dfact(j1 + j2 - j3) * dfact(j1 - j2 + j3) *
                      dfact(-j1 + j2 + j3) / dfact(j1 + j2 + j3 + 1));
    pre *= sqrt(dfact(j1 + m1) * dfact(j1 - m1) * dfact(j2 + m2) * dfact(j2 - m2) *
                dfact(j3 + m3) * dfact(j3 - m3));
    int kmin = 0;
    if (j2 - j3 - m1 > kmin) kmin = j2 - j3 - m1;
    if (j1 + m2 - j3 > kmin) kmin = j1 + m2 - j3;
    int kmax = j1 + j2 - j3;
    if (j1 - m1 < kmax) kmax = j1 - m1;
    if (j2 + m2 < kmax) kmax = j2 + m2;
    double s = 0.0;
    for (int k = kmin; k <= kmax; ++k) {
        double den = dfact(k) * dfact(j1 + j2 - j3 - k) * dfact(j1 - m1 - k) *
                     dfact(j2 + m2 - k) * dfact(j3 - j2 + m1 + k) * dfact(j3 - j1 - m2 + k);
        s += ((k & 1) ? -1.0 : 1.0) / den;
    }
    return pre * s;
}

__device__ void calc_q(int l, dc q[5][5])
{
    for (int i = 0; i < 5; ++i) for (int j = 0; j < 5; ++j) q[i][j] = {0.0, 0.0};
    const double s = 0.7071067811865476;
    for (int m = -l; m < 0; ++m) { q[l + m][l - m] = {s, 0.0}; q[l + m][l + m] = {0.0, -s}; }
    q[l][l] = {1.0, 0.0};
    for (int m = 1; m <= l; ++m) {
        double sg = (m & 1) ? -1.0 : 1.0;
        q[l + m][l + m] = {sg * s, 0.0};
        q[l + m][l - m] = {0.0, sg * s};
    }
    dc ph = {1.0, 0.0};                  // (-i)^l
    if (l == 1) ph = {0.0, -1.0};
    else if (l == 2) ph = {-1.0, 0.0};
    for (int i = 0; i < 5; ++i) for (int j = 0; j < 5; ++j) q[i][j] = cmul(ph, q[i][j]);
}

__global__ void k_w3j(float* __restrict__ out)
{
    if (threadIdx.x != 0 || blockIdx.x != 0) return;
    const int L1[15] = {0,0,0,1,1,1,1,1,1,2,2,2,2,2,2};
    const int L2[15] = {0,1,2,0,1,1,1,2,2,0,1,1,2,2,2};
    const int L3[15] = {0,1,2,1,0,1,2,1,2,2,1,2,0,1,2};
    const int OFF[15] = {0,1,10,35,44,53,80,125,170,245,270,315,390,415,490};
    for (int p = 0; p < 15; ++p) {
        int l1 = L1[p], l2 = L2[p], l3 = L3[p];
        int n1 = 2 * l1 + 1, n2 = 2 * l2 + 1, n3 = 2 * l3 + 1;
        dc q1[5][5], q2[5][5], q3[5][5];
        calc_q(l1, q1); calc_q(l2, q2); calc_q(l3, q3);
        double T[5][5][5];
        for (int a = 0; a < 5; ++a) for (int b = 0; b < 5; ++b) for (int c = 0; c < 5; ++c) T[a][b][c] = 0.0;
        for (int m1 = -l1; m1 <= l1; ++m1)
            for (int m2 = -l2; m2 <= l2; ++m2) {
                int m3 = -(m1 + m2);
                if (m3 < -l3 || m3 > l3) continue;
                double ph = (((l1 - l2 - m3) & 1) ? -1.0 : 1.0);
                T[m1 + l1][m2 + l2][m3 + l3] =
                    ph / sqrt((double)(2 * l3 + 1)) * cgc(l1, m1, l2, m2, l3, m1 + m2);
            }
        double Cre[5][5][5], Cim[5][5][5];
        double rmax = 0.0, imax = 0.0;
        for (int i = 0; i < n1; ++i) for (int j = 0; j < n2; ++j) for (int k = 0; k < n3; ++k) {
            double are = 0.0, aim = 0.0;
            for (int a = 0; a < n1; ++a) for (int b = 0; b < n2; ++b) for (int c = 0; c < n3; ++c) {
                double tv = T[a][b][c];
                if (tv == 0.0) continue;
                dc pr = cmul(cmul(q1[i][a], q2[j][b]), cconj(q3[k][c]));
                are += pr.re * tv; aim += pr.im * tv;
            }
            Cre[i][j][k] = are; Cim[i][j][k] = aim;
            if (fabs(are) > rmax) rmax = fabs(are);
            if (fabs(aim) > imax) imax = fabs(aim);
        }
        int useRe = (rmax >= imax);
        double nrm = 0.0;
        for (int i = 0; i < n1; ++i) for (int j = 0; j < n2; ++j) for (int k = 0; k < n3; ++k) {
            double w = useRe ? Cre[i][j][k] : Cim[i][j][k];
            nrm += w * w;
        }
        nrm = sqrt(nrm);
        double inv = (nrm > 0.0) ? 1.0 / nrm : 0.0;
        for (int i = 0; i < n1; ++i) for (int j = 0; j < n2; ++j) for (int k = 0; k < n3; ++k) {
            double w = useRe ? Cre[i][j][k] : Cim[i][j][k];
            out[OFF[p] + (i * n2 + j) * n3 + k] = (float)(w * inv);
        }
    }
}

// ---------------------------------------------------------------------------
// Message path tables (msgb = l3-section base {0,128,800} + mu_offset*(2l3+1))
// ---------------------------------------------------------------------------
static const MTable TAB0 = {3, 24, {
    {8, 0, 1, 0, 1, 1,  0,   0},   // (0,0,0)
    {8, 0, 1, 1, 3, 3,  1, 128},   // (0,1,1)
    {8, 0, 1, 4, 5, 5, 10, 800},   // (0,2,2)
}};

static const MTable TABN = {15, 576, {
    {64,   0, 1, 0, 1, 1,   0,    0},  // (0,0,0)
    {64,   0, 1, 1, 3, 3,   1,  128},  // (0,1,1)
    {64,   0, 1, 4, 5, 5,  10,  800},  // (0,2,2)
    {32,  64, 3, 0, 1, 3,  35,  320},  // (1,0,1)
    {32,  64, 3, 1, 3, 1,  44,   64},  // (1,1,0)
    {32,  64, 3, 1, 3, 3,  53,  416},  // (1,1,1)
    {32,  64, 3, 1, 3, 5,  80, 1120},  // (1,1,2)
    {32,  64, 3, 4, 5, 3, 125,  512},  // (1,2,1)
    {32,  64, 3, 4, 5, 5, 170, 1280},  // (1,2,2)
    {32, 160, 5, 0, 1, 5, 245, 1440},  // (2,0,2)
    {32, 160, 5, 1, 3, 3, 270,  608},  // (2,1,1)
    {32, 160, 5, 1, 3, 5, 315, 1600},  // (2,1,2)
    {32, 160, 5, 4, 5, 1, 390,   96},  // (2,2,0)
    {32, 160, 5, 4, 5, 3, 415,  704},  // (2,2,1)
    {32, 160, 5, 4, 5, 5, 490, 1760},  // (2,2,2)
}};

// ---------------------------------------------------------------------------
// Host launch
// ---------------------------------------------------------------------------
static inline void gemm(hipStream_t s, const float* A, int arow, int acol,
                        const float* B, float* C, int crow, int ccol,
                        int M, int K, int N, float scale, int act, int accum)
{
    dim3 g((N + 15) / 16, (M + 15) / 16);
    if (acol == 1)
        k_gemm_wmma<1><<<g, 32, 0, s>>>(A, arow, acol, B, C, crow, ccol, M, K, N, scale, act, accum);
    else
        k_gemm_wmma<0><<<g, 32, 0, s>>>(A, arow, acol, B, C, crow, ccol, M, K, N, scale, act, accum);
}

extern "C" void kernel_launch(void* const* d_in, const int* in_sizes, int n_in,
                              void* d_out, int out_size, void* d_ws, size_t ws_size,
                              hipStream_t stream)
{
    (void)in_sizes; (void)n_in; (void)out_size; (void)ws_size;

    const float* h_node_x = (const float*)d_in[0];
    const float* h_node_z = (const float*)d_in[1];
    const float* h_edge   = (const float*)d_in[2];
    const float* edge_vec = (const float*)d_in[3];
    const float* t_in     = (const float*)d_in[4];
    const float* y_in     = (const float*)d_in[5];
    const int*   edge_idx = (const int*)d_in[6];
    const int*   batch    = (const int*)d_in[7];

    struct LP { const float* sc[3]; const float* lin1[3]; const float* fc[3]; const float* lin2[3]; };
    LP L[3];
    int pi = 8;
    // layer 0 (in_muls = {0:8})
    L[0].sc[0]   = (const float*)d_in[pi++]; L[0].sc[1] = nullptr; L[0].sc[2] = nullptr;
    L[0].lin1[0] = (const float*)d_in[pi++]; L[0].lin1[1] = nullptr; L[0].lin1[2] = nullptr;
    for (int i = 0; i < 3; ++i) L[0].fc[i]   = (const float*)d_in[pi++];
    for (int i = 0; i < 3; ++i) L[0].lin2[i] = (const float*)d_in[pi++];
    // layers 1,2 (in_muls = {0:64,1:32,2:32})
    for (int ly = 1; ly < 3; ++ly) {
        for (int i = 0; i < 3; ++i) L[ly].sc[i]   = (const float*)d_in[pi++];
        for (int i = 0; i < 3; ++i) L[ly].lin1[i] = (const float*)d_in[pi++];
        for (int i = 0; i < 3; ++i) L[ly].fc[i]   = (const float*)d_in[pi++];
        for (int i = 0; i < 3; ++i) L[ly].lin2[i] = (const float*)d_in[pi++];
    }
    const float* tw1 = (const float*)d_in[pi++]; const float* tb1 = (const float*)d_in[pi++];
    const float* tw2 = (const float*)d_in[pi++]; const float* tb2 = (const float*)d_in[pi++];
    const float* pw1 = (const float*)d_in[pi++]; const float* pb1 = (const float*)d_in[pi++];
    const float* pw2 = (const float*)d_in[pi++]; const float* pb2 = (const float*)d_in[pi++];
    const float* yw1 = (const float*)d_in[pi++]; const float* yb1 = (const float*)d_in[pi++];
    const float* yw2 = (const float*)d_in[pi++]; const float* yb2 = (const float*)d_in[pi++];
    const float* qw1 = (const float*)d_in[pi++]; const float* qb1 = (const float*)d_in[pi++];
    const float* qw2 = (const float*)d_in[pi++]; const float* qb2 = (const float*)d_in[pi++];
    const float* outW = (const float*)d_in[pi++];

    // workspace carve-up (floats)
    float* W    = (float*)d_ws;
    float* w3j  = W;                       // 1024
    float* hty  = w3j  + 1024;             // 4*320
    float* sh   = hty  + 1280;             // 49152*9
    float* feat = sh   + (size_t)N_EDGES * 9;     // 4096*320
    float* x1   = feat + (size_t)N_ATOMS * 320;   // 4096*320
    float* fnew = x1   + (size_t)N_ATOMS * 320;   // 4096*384
    float* msg  = fnew + (size_t)N_ATOMS * 384;   // 4096*1920 (contiguous w/ fnew)
    float* Pbuf = msg  + (size_t)N_ATOMS * 1920;  // 4096*512
    float* r1   = Pbuf + (size_t)N_ATOMS * 512;   // 49152*16
    float* r2   = r1   + (size_t)N_EDGES * 16;    // 49152*64
    float* wbuf = r2   + (size_t)N_EDGES * 64;    // 49152*576

    k_w3j<<<1, 32, 0, stream>>>(w3j);
    k_embed<<<1, 320, 0, stream>>>(t_in, y_in,
                                   tw1, tb1, tw2, tb2, pw1, pb1, pw2, pb2,
                                   yw1, yb1, yw2, yb2, qw1, qb1, qw2, qb2, hty);
    k_sph<<<(N_EDGES + 255) / 256, 256, 0, stream>>>(edge_vec, sh);

    const int muls[3][3]  = {{8, 0, 0}, {64, 32, 32}, {64, 32, 32}};
    const int xbaseN[3]   = {0, 64, 160};
    const int outmul[3]   = {128, 32, 32};
    const int fnbase[3]   = {0, 128, 224};
    const int midK[3][3]  = {{8, 8, 8}, {128, 224, 224}, {128, 224, 224}};
    const int secbase[3]  = {0, 128, 800};
    const int nws[3]      = {24, 576, 576};

    for (int ly = 0; ly < 3; ++ly) {
        const float* fin = (ly == 0) ? h_node_x : feat;
        int frow = (ly == 0) ? 8 : 320;
        const int* mul = muls[ly];
        int nw = nws[ly];

        // zero fnew + msg (contiguous region)
        {
            long n = (long)N_ATOMS * (384 + 1920);
            k_fill<<<(int)((n + 255) / 256), 256, 0, stream>>>(fnew, n);
        }

        // x1 = lin1(feat) / sqrt(mul)  (per l, per m)
        for (int l = 0; l < 3; ++l) {
            if (!mul[l]) continue;
            int dm = 2 * l + 1;
            float sc = 1.0f / sqrtf((float)mul[l]);
            for (int m = 0; m < dm; ++m)
                gemm(stream, fin + xbaseN[l] + m, frow, dm, L[ly].lin1[l],
                     x1 + xbaseN[l] + m, 320, dm,
                     N_ATOMS, mul[l], mul[l], sc, 0, 0);
        }

        // sc self-connection: fnew += (x ⊗ z) @ W / sqrt(mul*8)
        for (int l = 0; l < 3; ++l) {
            if (!mul[l]) continue;
            int dm = 2 * l + 1;
            int K = mul[l] * 8;
            float sc = 1.0f / sqrtf((float)K);
            for (int m = 0; m < dm; ++m) {
                long tot = (long)N_ATOMS * K;
                k_outer<<<(int)((tot + 255) / 256), 256, 0, stream>>>(
                    fin, frow, xbaseN[l], dm, m, h_node_z, Pbuf, mul[l]);
                gemm(stream, Pbuf, K, 1, L[ly].sc[l],
                     fnew + fnbase[l] + m, 384, dm,
                     N_ATOMS, K, outmul[l], sc, 0, 1);
            }
        }

        // radial MLP on edges (SiLU, SiLU, linear)
        gemm(stream, h_edge, 8, 1, L[ly].fc[0], r1, 16, 1,
             N_EDGES, 8, 16, 1.0f / sqrtf(8.f), 1, 0);
        gemm(stream, r1, 16, 1, L[ly].fc[1], r2, 64, 1,
             N_EDGES, 16, 64, 0.25f, 1, 0);
        gemm(stream, r2, 64, 1, L[ly].fc[2], wbuf, nw, 1,
             N_EDGES, 64, nw, 0.125f, 0, 0);

        // tensor-product messages + scatter into msg
        {
            MTable T = (ly == 0) ? TAB0 : TABN;
            long tot = (long)N_EDGES * nw;
            k_msg<<<(int)((tot + 255) / 256), 256, 0, stream>>>(
                edge_idx, x1, sh, wbuf, w3j, msg, T);
        }

        // lin2: fnew += msg @ W / sqrt(mid)
        for (int l3 = 0; l3 < 3; ++l3) {
            int dm = 2 * l3 + 1;
            int K = midK[ly][l3];
            float sc = 1.0f / sqrtf((float)K);
            for (int m = 0; m < dm; ++m)
                gemm(stream, msg + secbase[l3] + m, 1920, dm, L[ly].lin2[l3],
                     fnew + fnbase[l3] + m, 384, dm,
                     N_ATOMS, K, outmul[l3], sc, 0, 1);
        }

        // gate + t/y conditioning -> feat for next layer
        {
            long tot = (long)N_ATOMS * 320;
            k_gate<<<(int)((tot + 255) / 256), 256, 0, stream>>>(fnew, hty, batch, feat);
        }
    }

    k_final<<<(N_ATOMS * 3 + 255) / 256, 256, 0, stream>>>(
        feat, h_node_z, outW, (float*)d_out);
}